// Pooling_net_76802605187717
// MI455X (gfx1250) — compile-verified
//
#include <hip/hip_runtime.h>

// CDNA5 / gfx1250, wave32. Uses V_WMMA_F32_16X16X4_F32.
typedef float v2f __attribute__((ext_vector_type(2)));
typedef float v8f __attribute__((ext_vector_type(8)));

#define NN_DEG 32
#define NN_A   65
#define NN_EMB 32
#define NN_HID 16
#define NN_BOT 32

// ---------------------------------------------------------------------------
// Prep: fold layer1+layer2 ->  Wa(2x16) = W_se(2x32) @ W1(32x16),
//                              ba(16)   = b_se(32) @ W1 + b1
// ws layout (floats): [0..15] Wa row0, [16..31] Wa row1, [32..47] ba
// ---------------------------------------------------------------------------
__global__ void prep_kernel(const float* __restrict__ W_se,
                            const float* __restrict__ b_se,
                            const float* __restrict__ W1,
                            const float* __restrict__ b1,
                            float* __restrict__ ws) {
  const int t = threadIdx.x;
  if (t < 32) {
    const int r = t >> 4, c = t & 15;
    float s = 0.f;
#pragma unroll
    for (int k = 0; k < NN_EMB; ++k) s += W_se[r * NN_EMB + k] * W1[k * NN_HID + c];
    ws[r * 16 + c] = s;
  }
  if (t < 16) {
    float s = b1[t];
#pragma unroll
    for (int k = 0; k < NN_EMB; ++k) s += b_se[k] * W1[k * NN_HID + t];
    ws[32 + t] = s;
  }
}

// ---------------------------------------------------------------------------
// Main kernel: one wave32 per node.
//   rows d=0..31 of node: rel = corr[node, nei1[node*32+d]]  (float2 gather)
//   h1^T(16x16) = Wa^T(16x4pad) x rel^T(4x16)   : 1 WMMA per 16-row tile
//   h2(16x32)   = relu(h1) @ W2 + b2            : 8 chained WMMAs per tile
//   out[node,:] = max over 32 rows of relu(h2)  (in-wave reduction)
// ---------------------------------------------------------------------------
__global__ __launch_bounds__(256) void pool_kernel(
    const float* __restrict__ corr,   // (N, 65, 2) f32
    const int*   __restrict__ nei,    // (N*32, 3) i32
    const float* __restrict__ ws,     // Wa(2x16), ba(16)
    const float* __restrict__ W2,     // (16, 32) f32
    const float* __restrict__ b2,     // (32) f32
    float* __restrict__ out,          // (N, 32) f32
    int N) {
  const int lane = threadIdx.x & 31;
  const int wv   = threadIdx.x >> 5;
  const int node = blockIdx.x * 8 + wv;
  if (node >= N) return;                 // wave-uniform: EXEC stays all-ones
  const int  lo = lane & 15;
  const bool hi = (lane >= 16);

  // ---- gather: data row d = lane of this node --------------------------------
  const int    a   = nei[(node * NN_DEG + lane) * 3 + 1];
  const float2 rel = ((const float2*)corr)[node * NN_A + a];
  const float rx  = rel.x, ry = rel.y;
  const float rxs = __shfl_xor(rx, 16, 32);   // other half's row (lo <-> lo+16)
  const float rys = __shfl_xor(ry, 16, 32);

  // ---- constant operands -----------------------------------------------------
  // A = Wa^T in 16x4 f32 A-format: slot (p, lane) -> (M=lo, K=p+2*hi); K>=2 pad 0
  v2f aW;
  aW[0] = hi ? 0.f : ws[lo];
  aW[1] = hi ? 0.f : ws[16 + lo];
  // C bias for transposed L1 product: C[M=j+8*hi][N] = ba[j+8*hi]
  float cba[8];
#pragma unroll
  for (int j = 0; j < 8; ++j) cba[j] = hi ? ws[32 + 8 + j] : ws[32 + j];
  // W2 in B-format chunks: slot (p, lane) of chunk k0c, tile n
  //   -> W2[K = 4*k0c + p + 2*hi][16*n + lo]
  float bw[4][2][2];
#pragma unroll
  for (int k0c = 0; k0c < 4; ++k0c)
#pragma unroll
    for (int p = 0; p < 2; ++p) {
      const int krow = k0c * 4 + p + (hi ? 2 : 0);
#pragma unroll
      for (int n = 0; n < 2; ++n) bw[k0c][p][n] = W2[krow * NN_BOT + n * 16 + lo];
    }
  const float cb2[2] = { b2[lo], b2[16 + lo] };

  float mx[2];                                   // per-N-tile column maxima
  // ---- two 16-row tiles per node --------------------------------------------
#pragma unroll
  for (int t = 0; t < 2; ++t) {
    // B = rel^T (4x16, K padded): slot (p,lane) -> rel[row=lo+16t][K=p+2*hi]
    v2f bR;
    bR[0] = hi ? 0.f : (t ? rxs : rx);
    bR[1] = hi ? 0.f : (t ? rys : ry);
    v8f c;
#pragma unroll
    for (int j = 0; j < 8; ++j) c[j] = cba[j];

    // h1^T = Wa^T x rel^T + ba   (D slot (j,lane) = h1[row=lo][feat=j+8*hi])
    v8f h1t = __builtin_amdgcn_wmma_f32_16x16x4_f32(
        false, aW, false, bR, (short)0, c, false, false);
#pragma unroll
    for (int j = 0; j < 8; ++j) h1t[j] = fmaxf(h1t[j], 0.f);   // ReLU

    // Re-slot h1 into 16x4 A-format chunks via cross-half swaps:
    //   need A_p = h1[lo][k0 + p + 2*hi]; have h1[lo][j + 8*hi] in h1t[j]
    v2f ach[4];
#pragma unroll
    for (int p = 0; p < 2; ++p) {
      float s0 = __shfl_xor(h1t[2 + p], 16, 32);
      ach[0][p] = hi ? s0 : h1t[0 + p];          // k0 = 0
      float s1 = __shfl_xor(h1t[6 + p], 16, 32);
      ach[1][p] = hi ? s1 : h1t[4 + p];          // k0 = 4
      float s2 = __shfl_xor(h1t[0 + p], 16, 32);
      ach[2][p] = hi ? h1t[2 + p] : s2;          // k0 = 8
      float s3 = __shfl_xor(h1t[4 + p], 16, 32);
      ach[3][p] = hi ? h1t[6 + p] : s3;          // k0 = 12
    }

    // h2 tile-n = sum_k0 A(k0) x W2(k0, n) + b2  -- 4 chained WMMAs each
#pragma unroll
    for (int n = 0; n < 2; ++n) {
      v8f acc;
#pragma unroll
      for (int j = 0; j < 8; ++j) acc[j] = cb2[n];
#pragma unroll
      for (int k0c = 0; k0c < 4; ++k0c) {
        v2f bb;
        bb[0] = bw[k0c][0][n];
        bb[1] = bw[k0c][1][n];
        acc = __builtin_amdgcn_wmma_f32_16x16x4_f32(
            false, ach[k0c], false, bb, (short)0, acc, false, false);
      }
      // ReLU + row-max fold: max over 8 rows in this lane, clamp at 0
      float m = acc[0];
#pragma unroll
      for (int j = 1; j < 8; ++j) m = fmaxf(m, acc[j]);
      m = fmaxf(m, 0.f);
      mx[n] = (t == 0) ? m : fmaxf(mx[n], m);
    }
  }

  // combine M=0..7 half with M=8..15 half (lane L <-> L^16), then store.
#pragma unroll
  for (int n = 0; n < 2; ++n) mx[n] = fmaxf(mx[n], __shfl_xor(mx[n], 16, 32));
  const float outv = hi ? mx[1] : mx[0];
  out[node * NN_BOT + (hi ? 16 : 0) + lo] = outv;   // coalesced 128B per wave
}

// ---------------------------------------------------------------------------
extern "C" void kernel_launch(void* const* d_in, const int* in_sizes, int n_in,
                              void* d_out, int out_size, void* d_ws, size_t ws_size,
                              hipStream_t stream) {
  const float* corr = (const float*)d_in[0];  // (N,65,2)
  const int*   nei  = (const int*)  d_in[1];  // (N*32,3)
  // d_in[2] = lstm_state (unused by reference)
  const float* W_se = (const float*)d_in[3];
  const float* b_se = (const float*)d_in[4];
  const float* W1   = (const float*)d_in[5];
  const float* b1   = (const float*)d_in[6];
  const float* W2   = (const float*)d_in[7];
  const float* b2   = (const float*)d_in[8];
  float* out = (float*)d_out;
  float* ws  = (float*)d_ws;

  const int N = in_sizes[0] / (NN_A * 2);

  prep_kernel<<<1, 64, 0, stream>>>(W_se, b_se, W1, b1, ws);

  const int waves_per_block = 8;                   // 256 threads
  const int blocks = (N + waves_per_block - 1) / waves_per_block;
  pool_kernel<<<blocks, 256, 0, stream>>>(corr, nei, ws, W2, b2, out, N);
}